// DglGraphConvolution_21715354648942
// MI455X (gfx1250) — compile-verified
//
#include <hip/hip_runtime.h>

typedef __attribute__((ext_vector_type(2))) float v2f;
typedef __attribute__((ext_vector_type(8))) float v8f;

constexpr int Bn  = 16;
constexpr int Nn  = 4096;
constexpr int En  = 131072;   // 2^17
constexpr int Fin = 128;
constexpr int Fout = 128;

// ---------------------------------------------------------------------------
// Kernel 1: hidden[M,128] = text[M,128] @ weight[128,128]   (M = B*N = 65536)
// 8 waves / block; wave w computes M-tile (blockIdx.x*8+w) x full 128 cols
// via V_WMMA_F32_16X16X4_F32, K swept in steps of 4.
// ---------------------------------------------------------------------------
__global__ __launch_bounds__(256) void gcn_gemm(const float* __restrict__ text,
                                                const float* __restrict__ weight,
                                                float* __restrict__ hidden) {
  __shared__ float w[Fin * Fout];          // 64 KB, row-major [K][N], same as global

  const int tid = threadIdx.x;
  // Cooperative 64 KB weight stage: coalesced, 64 floats per thread.
#pragma unroll
  for (int i = 0; i < (Fin * Fout) / 256; ++i) {
    const int idx = tid + i * 256;
    w[idx] = weight[idx];
  }
  __syncthreads();

  const int wave    = tid >> 5;
  const int lane    = tid & 31;
  const int halfsel = lane >> 4;           // 0: K pair {0,1}, 1: K pair {2,3}
  const int l16     = lane & 15;
  const int koff    = halfsel * 2;

  const long mtile = (long)blockIdx.x * 8 + wave;
  const float* arow = text + (mtile * 16 + l16) * (long)Fin;  // A: M = l16 in both halves

  v8f acc[8] = {};

  for (int k0 = 0; k0 < Fin; k0 += 4) {
    // A fragment 16x4 f32: lane holds A[M=l16][k0+koff], A[M=l16][k0+koff+1]
    v2f a;
    a[0] = arow[k0 + koff];
    a[1] = arow[k0 + koff + 1];
#pragma unroll
    for (int nt = 0; nt < 8; ++nt) {
      // B fragment 4x16 f32: lane holds B[k0+koff][n], B[k0+koff+1][n], n = nt*16+l16
      v2f b;
      b[0] = w[(k0 + koff) * Fout + nt * 16 + l16];
      b[1] = w[(k0 + koff + 1) * Fout + nt * 16 + l16];
      acc[nt] = __builtin_amdgcn_wmma_f32_16x16x4_f32(
          /*neg_a=*/false, a, /*neg_b=*/false, b,
          /*c_mod=*/(short)0, acc[nt], /*reuse_a=*/false, /*reuse_b=*/false);
    }
  }

  // C/D layout: VGPR r -> (M=r, N=lane) for lanes 0-15; (M=r+8, N=lane-16) for 16-31
  const long rowBase = mtile * 16 + halfsel * 8;
#pragma unroll
  for (int nt = 0; nt < 8; ++nt) {
#pragma unroll
    for (int r = 0; r < 8; ++r) {
      hidden[(rowBase + r) * Fout + nt * 16 + l16] = acc[nt][r];
    }
  }
}

// ---------------------------------------------------------------------------
// Kernel 2: zero agg + deg scratch (float4 stores)
// ---------------------------------------------------------------------------
__global__ __launch_bounds__(256) void gcn_zero(float* __restrict__ p, long n) {
  const long idx = ((long)blockIdx.x * 256 + threadIdx.x) * 4;
  if (idx < n) {
    float4 z = {0.f, 0.f, 0.f, 0.f};
    *reinterpret_cast<float4*>(p + idx) = z;
  }
}

// ---------------------------------------------------------------------------
// Kernel 3: per-edge scatter-add. 128 threads per edge (one per feature):
// coalesced 512B gather of hidden[src], global_atomic_add_f32 into agg[dst].
// Lane f==0 also counts the edge into deg[dst].
// ---------------------------------------------------------------------------
__global__ __launch_bounds__(256) void gcn_edge(const float* __restrict__ hidden,
                                                const int* __restrict__ esrc,
                                                const int* __restrict__ edst,
                                                float* __restrict__ agg,
                                                float* __restrict__ deg) {
  const long idx = (long)blockIdx.x * 256 + threadIdx.x;
  const int  f   = (int)(idx & (Fout - 1));
  const long eg  = idx >> 7;                 // global edge id, 0..B*E-1
  const int  b   = (int)(eg >> 17);          // E = 2^17
  const int  e   = (int)(eg & (En - 1));
  const long ei  = (long)b * En + e;

  const int src = esrc[ei];
  const int dst = edst[ei];

  const float val = hidden[((long)b * Nn + src) * Fout + f];
  atomicAdd(&agg[((long)b * Nn + dst) * Fout + f], val);
  if (f == 0) {
    atomicAdd(&deg[(long)b * Nn + dst], 1.0f);
  }
}

// ---------------------------------------------------------------------------
// Kernel 4: out = agg / (deg + 1) + bias
// ---------------------------------------------------------------------------
__global__ __launch_bounds__(256) void gcn_final(const float* __restrict__ agg,
                                                 const float* __restrict__ deg,
                                                 const float* __restrict__ bias,
                                                 float* __restrict__ out) {
  const long idx  = (long)blockIdx.x * 256 + threadIdx.x;
  const int  f    = (int)(idx & (Fout - 1));
  const long node = idx >> 7;
  out[idx] = agg[idx] / (deg[node] + 1.0f) + bias[f];
}

// ---------------------------------------------------------------------------
extern "C" void kernel_launch(void* const* d_in, const int* in_sizes, int n_in,
                              void* d_out, int out_size, void* d_ws, size_t ws_size,
                              hipStream_t stream) {
  const float* text   = (const float*)d_in[0];
  const float* weight = (const float*)d_in[1];
  const float* bias   = (const float*)d_in[2];
  const int*   esrc   = (const int*)d_in[3];
  const int*   edst   = (const int*)d_in[4];
  float* out = (float*)d_out;

  float* agg = (float*)d_ws;                          // B*N*Fout floats
  float* deg = agg + (long)Bn * Nn * Fout;            // B*N floats

  // 1) GEMM: hidden staged in d_out
  gcn_gemm<<<(Bn * Nn) / (16 * 8), 256, 0, stream>>>(text, weight, out);

  // 2) Zero agg + deg
  const long zn = (long)Bn * Nn * Fout + (long)Bn * Nn;   // multiple of 4
  gcn_zero<<<(int)((zn / 4 + 255) / 256), 256, 0, stream>>>(agg, zn);

  // 3) Edge scatter-add
  const long ethreads = (long)Bn * En * Fout;
  gcn_edge<<<(int)(ethreads / 256), 256, 0, stream>>>(out, esrc, edst, agg, deg);

  // 4) Normalize + bias -> d_out
  gcn_final<<<(Bn * Nn * Fout) / 256, 256, 0, stream>>>(agg, deg, bias, out);
}